// GraphAttentionLayer_86835648790656
// MI455X (gfx1250) — compile-verified
//
#include <hip/hip_runtime.h>
#include <math.h>

#define N_NODES 100000
#define N_EDGES 3200000
#define F_IN    256
#define D_OUT   128

typedef float v2f __attribute__((ext_vector_type(2)));
typedef float v8f __attribute__((ext_vector_type(8)));

// ---------------------------------------------------------------------------
// K0: fold W1@w2 -> u1[256], W1@w3 -> u2[256]   (tiny, one block)
// ---------------------------------------------------------------------------
__global__ void k_fold_w(const float* __restrict__ W1,
                         const float* __restrict__ w2,
                         const float* __restrict__ w3,
                         float* __restrict__ u1, float* __restrict__ u2) {
    int k = threadIdx.x;                 // 0..255
    if (k < F_IN) {
        float s1 = 0.f, s2 = 0.f;
        for (int d = 0; d < D_OUT; ++d) {
            float w = W1[(size_t)k * D_OUT + d];
            s1 += w * w2[d];
            s2 += w * w3[d];
        }
        u1[k] = s1;
        u2[k] = s2;
    }
}

// ---------------------------------------------------------------------------
// K1: sa1[n] = x[n,:]·u1 + b2 ; sa2[n] = x[n,:]·u2 + b3   (one wave32/node)
// ---------------------------------------------------------------------------
__global__ void __launch_bounds__(256)
k_sa(const float* __restrict__ x,
     const float* __restrict__ u1, const float* __restrict__ u2,
     const float* __restrict__ b2, const float* __restrict__ b3,
     float* __restrict__ sa1, float* __restrict__ sa2) {
    int wave = threadIdx.x >> 5;
    int lane = threadIdx.x & 31;
    int n = blockIdx.x * 8 + wave;
    if (n >= N_NODES) return;
    const float* xr = x + (size_t)n * F_IN;
    float s1 = 0.f, s2 = 0.f;
    for (int j = lane; j < F_IN; j += 32) {
        float v = xr[j];
        s1 += v * u1[j];
        s2 += v * u2[j];
    }
    for (int off = 16; off > 0; off >>= 1) {
        s1 += __shfl_down(s1, off);
        s2 += __shfl_down(s2, off);
    }
    if (lane == 0) {
        sa1[n] = s1 + b2[0];
        sa2[n] = s2 + b3[0];
    }
}

// ---------------------------------------------------------------------------
// K2: value = x @ kernel  via V_WMMA_F32_16X16X4_F32 (fp32-exact matrix op).
// One wave per 16x16 C tile; 8 waves/block -> 16x128 strip; K-loop step 4.
// A 16x4: lanes0-15 M=lane,K={0,1}; lanes16-31 K={2,3}.
// B 4x16: VGPR0 K={0|2}, VGPR1 K={1|3}, N=lane&15.
// D: VGPR v -> row m0+v+8*half, col n0+(lane&15).
// ---------------------------------------------------------------------------
__global__ void __launch_bounds__(256)
k_value_wmma(const float* __restrict__ x,
             const float* __restrict__ Wk,
             float* __restrict__ value) {
    const int wave = threadIdx.x >> 5;
    const int lane = threadIdx.x & 31;
    const int half = lane >> 4;
    const int lrow = lane & 15;
    const int m0 = blockIdx.x * 16;
    const int n0 = wave * 16;

    const float* arow = x + (size_t)(m0 + lrow) * F_IN + 2 * half;
    const float* bcol = Wk + (size_t)(2 * half) * D_OUT + (n0 + lrow);

    v8f c = {0.f, 0.f, 0.f, 0.f, 0.f, 0.f, 0.f, 0.f};
    #pragma unroll 8
    for (int k = 0; k < F_IN; k += 4) {
        v2f a = *(const v2f*)(arow + k);              // K = k+2h, k+2h+1
        v2f b;
        b.x = bcol[(size_t)k * D_OUT];                // K = k+2h
        b.y = bcol[(size_t)(k + 1) * D_OUT];          // K = k+2h+1
        c = __builtin_amdgcn_wmma_f32_16x16x4_f32(
                /*neg_a=*/false, a, /*neg_b=*/false, b,
                /*c_mod=*/(short)0, c, /*reuse_a=*/false, /*reuse_b=*/false);
    }

    float* orow = value + (size_t)(m0 + 8 * half) * D_OUT + n0 + lrow;
    #pragma unroll
    for (int v = 0; v < 8; ++v)
        orow[(size_t)v * D_OUT] = c[v];
}

// ---------------------------------------------------------------------------
// K3: fused edge-score + row softmax + SpMM + bias. One 128-thread block per
// row; edge range found by binary search over sorted edge_row. Deterministic
// fixed-tree reductions (no float atomics). Empty rows -> out = bias.
// ---------------------------------------------------------------------------
__global__ void __launch_bounds__(128)
k_gat_spmm(const float* __restrict__ adj,
           const int* __restrict__ erow,
           const int* __restrict__ ecol,
           const float* __restrict__ sa1,
           const float* __restrict__ sa2,
           const float* __restrict__ value,
           const float* __restrict__ bias,
           float* __restrict__ out) {
    __shared__ int   s_start, s_end;
    __shared__ float red[128];
    __shared__ float satt[128];
    __shared__ int   scol[128];

    const int r = blockIdx.x;
    const int tid = threadIdx.x;

    if (tid == 0) {
        int lo = 0, hi = N_EDGES;
        while (lo < hi) { int mid = (lo + hi) >> 1; if (erow[mid] <  r) lo = mid + 1; else hi = mid; }
        s_start = lo;
        int lo2 = lo; hi = N_EDGES;
        while (lo2 < hi) { int mid = (lo2 + hi) >> 1; if (erow[mid] <= r) lo2 = mid + 1; else hi = mid; }
        s_end = lo2;
    }
    __syncthreads();
    const int start = s_start, end = s_end;
    const float s1r = sa1[r];

    // pass 1: row max of leaky_relu(adj*(sa1[r]+sa2[col]))
    float mx = -INFINITY;
    for (int e = start + tid; e < end; e += 128) {
        float v = adj[e] * (s1r + sa2[ecol[e]]);
        v = v > 0.f ? v : 0.2f * v;
        mx = fmaxf(mx, v);
    }
    red[tid] = mx; __syncthreads();
    for (int s = 64; s > 0; s >>= 1) {
        if (tid < s) red[tid] = fmaxf(red[tid], red[tid + s]);
        __syncthreads();
    }
    const float m = red[0];
    __syncthreads();

    // pass 2: denom = sum exp(e - m)
    float ds = 0.f;
    for (int e = start + tid; e < end; e += 128) {
        float v = adj[e] * (s1r + sa2[ecol[e]]);
        v = v > 0.f ? v : 0.2f * v;
        ds += __expf(v - m);
    }
    red[tid] = ds; __syncthreads();
    for (int s = 64; s > 0; s >>= 1) {
        if (tid < s) red[tid] += red[tid + s];
        __syncthreads();
    }
    const float denom = red[0];
    const float dinv = denom > 0.f ? 1.0f / denom : 0.0f;
    __syncthreads();

    // pass 3: out[r, tid] = bias + sum att[e] * value[col[e], tid]
    float acc = bias[(size_t)r * D_OUT + tid];
    for (int base = start; base < end; base += 128) {
        int idx = base + tid;
        if (idx < end) {
            int c = ecol[idx];
            float v = adj[idx] * (s1r + sa2[c]);
            v = v > 0.f ? v : 0.2f * v;
            satt[tid] = __expf(v - m) * dinv;
            scol[tid] = c;
        }
        __syncthreads();
        int lim = end - base; if (lim > 128) lim = 128;
        for (int j = 0; j < lim; ++j)
            acc += satt[j] * value[(size_t)scol[j] * D_OUT + tid];
        __syncthreads();
    }
    out[(size_t)r * D_OUT + tid] = acc;
}

// ---------------------------------------------------------------------------
extern "C" void kernel_launch(void* const* d_in, const int* in_sizes, int n_in,
                              void* d_out, int out_size, void* d_ws, size_t ws_size,
                              hipStream_t stream) {
    (void)in_sizes; (void)n_in; (void)out_size; (void)ws_size;

    const float* x    = (const float*)d_in[0];
    const float* adj  = (const float*)d_in[1];
    const float* W1   = (const float*)d_in[2];
    const float* w2   = (const float*)d_in[3];
    const float* b2   = (const float*)d_in[4];
    const float* w3   = (const float*)d_in[5];
    const float* b3   = (const float*)d_in[6];
    const float* Wk   = (const float*)d_in[7];
    const float* bias = (const float*)d_in[8];
    const int*   erow = (const int*)d_in[9];
    const int*   ecol = (const int*)d_in[10];
    float* out = (float*)d_out;

    // workspace layout (all 256B aligned)
    char* ws = (char*)d_ws;
    float* u1    = (float*)ws;                      ws += 256 * sizeof(float);
    float* u2    = (float*)ws;                      ws += 256 * sizeof(float);
    float* sa1   = (float*)ws;                      ws += ((N_NODES + 63) & ~63) * sizeof(float);
    float* sa2   = (float*)ws;                      ws += ((N_NODES + 63) & ~63) * sizeof(float);
    float* value = (float*)ws;                      // N_NODES * D_OUT floats (51.2 MB)

    k_fold_w<<<1, 256, 0, stream>>>(W1, w2, w3, u1, u2);
    k_sa<<<(N_NODES + 7) / 8, 256, 0, stream>>>(x, u1, u2, b2, b3, sa1, sa2);
    k_value_wmma<<<N_NODES / 16, 256, 0, stream>>>(x, Wk, value);
    k_gat_spmm<<<N_NODES, 128, 0, stream>>>(adj, erow, ecol, sa1, sa2, value, bias, out);
}